// PairwiseEnergy_87909390615159
// MI455X (gfx1250) — compile-verified
//
#include <hip/hip_runtime.h>
#include <math.h>

typedef __attribute__((ext_vector_type(2))) float v2f;
typedef __attribute__((ext_vector_type(8))) float v8f;

#define PE_BETA_MAX 10.0f

__device__ __forceinline__ float pe_softplus(float x) {
  // numerically stable softplus: log(1+e^x)
  return (x > 20.0f) ? x : log1pf(expf(x));
}

__device__ __forceinline__ v8f pe_wmma_acc(v8f acc, float2 qa, float2 kb) {
  v2f A; A.x = qa.x; A.y = qa.y;
  v2f B; B.x = kb.x; B.y = kb.y;
  return __builtin_amdgcn_wmma_f32_16x16x4_f32(
      /*neg_a=*/false, A, /*neg_b=*/false, B,
      /*c_mod=*/(short)0, acc, /*reuse_a=*/false, /*reuse_b=*/false);
}

// Fully-unrolled Gram tile for a compile-time chunk count: stage all loads
// first (compiler clauses them, single s_wait_loadcnt), then run the WMMA
// chain back-to-back so loads of the next head overlap this head's math.
template <int NK>
__device__ __forceinline__ v8f pe_gram(const float* __restrict__ qp,
                                       const float* __restrict__ kp,
                                       int khalf) {
  float2 qa[NK], kb[NK];
#pragma unroll
  for (int kc = 0; kc < NK; ++kc) {
    const int d0 = (kc << 2) + (khalf << 1);
    qa[kc] = *(const float2*)(qp + d0);
    kb[kc] = *(const float2*)(kp + d0);
  }
  v8f acc = {0.f, 0.f, 0.f, 0.f, 0.f, 0.f, 0.f, 0.f};
#pragma unroll
  for (int kc = 0; kc < NK; ++kc) acc = pe_wmma_acc(acc, qa[kc], kb[kc]);
  return acc;
}

// ---------------------------------------------------------------------------
// Edge pass: one wave handles a tile of 16 edges.
// Per head h: 16x16 Gram tile of gathered Q rows (A) vs K rows (B) via
// V_WMMA_F32_16X16X4_F32 over D/4 chunks; extract the diagonal (per-edge
// dots); atomicAdd(exp(beta*(dot/sqrt(D)+a_h))) into s_node[c*H+h].
//
// 32-bit WMMA operand layout (ISA 7.12.2):
//   A 16x4 : lane m (m<16) holds A[m][0],A[m][1] in V0,V1; lane m+16 K=2,3
//   B 4x16 : mirrored (column n on lane n, K split across lane halves)
// => both A and B load a float2 at  row_base + kc*4 + (lane>>4)*2.
//   D 16x16: VGPR v, lanes 0-15 -> M=v,N=lane; lanes 16-31 -> M=v+8,N=lane-16
// => diag(m): m<8 -> acc[m] @ lane m ; m>=8 -> acc[m-8] @ lane m+16.
// ---------------------------------------------------------------------------
__global__ void pe_edge_kernel(const float* __restrict__ Q2,
                               const float* __restrict__ K2,
                               const float* __restrict__ a2,
                               const float* __restrict__ beta_raw,
                               const int*   __restrict__ c2,
                               const int*   __restrict__ u2,
                               float*       __restrict__ s_node,
                               int E, int H, int D, float invScale) {
  const int lane = threadIdx.x & 31;
  const int wave = threadIdx.x >> 5;
  const int tile = blockIdx.x * (blockDim.x >> 5) + wave;
  const int base = tile << 4;
  if (base >= E) return;  // wave-uniform exit: EXEC stays all-ones for WMMA

  const float beta = fminf(pe_softplus(beta_raw[0]), PE_BETA_MAX);

  const int  m     = lane & 15;
  const int  khalf = lane >> 4;
  const int  e     = base + m;
  const bool valid = (e < E);
  const int  ec    = valid ? e : base;  // clamp tail lanes to a legal edge
  const int  cn    = c2[ec];
  const int  un    = u2[ec];

  for (int h = 0; h < H; ++h) {
    const float* qp = Q2 + ((size_t)cn * H + h) * D;
    const float* kp = K2 + ((size_t)un * H + h) * D;

    v8f acc;
    if (D == 32) {
      acc = pe_gram<8>(qp, kp, khalf);       // fast path: fully unrolled
    } else {                                  // generic D (multiple of 4)
      acc = (v8f){0.f, 0.f, 0.f, 0.f, 0.f, 0.f, 0.f, 0.f};
      const int nk = D >> 2;
      for (int kc = 0; kc < nk; ++kc) {
        const int d0 = (kc << 2) + (khalf << 1);
        acc = pe_wmma_acc(acc, *(const float2*)(qp + d0),
                               *(const float2*)(kp + d0));
      }
    }

    // Pull diag element for edge m onto lane m (lanes 16-31 get copies)
    float diag = 0.f;
#pragma unroll
    for (int v = 0; v < 8; ++v) {
      const float lo = __shfl(acc[v], v, 32);       // D[v][v]
      const float hi = __shfl(acc[v], v + 24, 32);  // D[v+8][v+8]
      diag = (m == v)     ? lo : diag;
      diag = (m == v + 8) ? hi : diag;
    }

    if (valid && lane < 16) {
      const float x = beta * (diag * invScale + a2[h]);
      atomicAdd(&s_node[(size_t)cn * H + h], expf(x));
    }
  }
}

// ---------------------------------------------------------------------------
// Node pass: lse = (s>0) ? log(s) : 0 ; out[batch[n]*H+h] += (lam/beta)*lse.
// batch is sorted, so each 256-thread block (64 nodes) touches ~1-2 graphs:
// stage in LDS, flush only nonzero cells to global atomics.
// ---------------------------------------------------------------------------
__global__ void pe_node_kernel(const float* __restrict__ s_node,
                               const int*   __restrict__ batch,
                               const float* __restrict__ lam_raw,
                               const float* __restrict__ beta_raw,
                               float*       __restrict__ out,
                               int N, int H, int outSize) {
  extern __shared__ float smem[];
  for (int i = threadIdx.x; i < outSize; i += blockDim.x) smem[i] = 0.f;
  __syncthreads();

  const int idx = blockIdx.x * blockDim.x + threadIdx.x;
  if (idx < N * H) {
    const int   n    = idx / H;
    const int   h    = idx - n * H;
    const float lam  = pe_softplus(lam_raw[0]);
    const float beta = fminf(pe_softplus(beta_raw[0]), PE_BETA_MAX);
    const float s    = s_node[idx];
    if (s > 0.f) {
      atomicAdd(&smem[batch[n] * H + h], (lam / beta) * logf(s));
    }
  }
  __syncthreads();
  for (int i = threadIdx.x; i < outSize; i += blockDim.x) {
    const float v = smem[i];
    if (v != 0.f) atomicAdd(&out[i], v);
  }
}

extern "C" void kernel_launch(void* const* d_in, const int* in_sizes, int n_in,
                              void* d_out, int out_size, void* d_ws, size_t ws_size,
                              hipStream_t stream) {
  // setup_inputs order: G, Q2, K2, a_2, lambda_2_raw, beta_2_raw,
  //                     c_2, u_2, batch, num_graphs, num_nodes
  const float* Q2       = (const float*)d_in[1];
  const float* K2       = (const float*)d_in[2];
  const float* a2       = (const float*)d_in[3];
  const float* lam_raw  = (const float*)d_in[4];
  const float* beta_raw = (const float*)d_in[5];
  const int*   c2       = (const int*)d_in[6];
  const int*   u2       = (const int*)d_in[7];
  const int*   batch    = (const int*)d_in[8];

  const int N = in_sizes[8];                    // batch has N entries
  const int H = in_sizes[3];                    // a_2 has H entries
  const int E = in_sizes[6];                    // c_2 has E entries
  const int D = (N > 0 && H > 0) ? (int)(in_sizes[1] / ((long)N * H)) : 0;

  float* s_node = (float*)d_ws;                 // N*H f32 accumulators (800 KB)

  hipMemsetAsync(s_node, 0, (size_t)N * H * sizeof(float), stream);
  hipMemsetAsync(d_out, 0, (size_t)out_size * sizeof(float), stream);

  if (E > 0 && N > 0 && H > 0 && D > 0) {
    const float invScale = 1.0f / sqrtf((float)D);
    const int tiles  = (E + 15) >> 4;
    const int wavesB = 8;  // 256 threads = 8 wave32 tiles per block
    const int blocks = (tiles + wavesB - 1) / wavesB;
    pe_edge_kernel<<<blocks, 256, 0, stream>>>(Q2, K2, a2, beta_raw, c2, u2,
                                               s_node, E, H, D, invScale);

    const int total   = N * H;
    const int nblocks = (total + 255) / 256;
    pe_node_kernel<<<nblocks, 256, (size_t)out_size * sizeof(float), stream>>>(
        s_node, batch, lam_raw, beta_raw, (float*)d_out, N, H, out_size);
  }
}